// HamiltonianLinearFlow_44650480009356
// MI455X (gfx1250) — compile-verified
//
#include <hip/hip_runtime.h>
#include <hip/hip_bf16.h>
#include <math.h>

// ---------------------------------------------------------------------------
// HamiltonianLinearFlow on MI455X (gfx1250)
//   out[t,n,:] = expm(tau(t,n) * JA) @ x[n],  tau = (1+0.01*tanh(x^T A x)) * t
// Since every expm is of ONE matrix JA scaled by a scalar:
//   expm(tau*JA) x = sum_k tau^k * U_k,  U_k = (JA)^k x / k!
// Krylov chain computed with V_WMMA_F32_16X16X4_F32 (full-precision f32 WMMA),
// 4 K=4 chunks per 16x16x16 product, one wave per 16-sample tile.
// ---------------------------------------------------------------------------

typedef __attribute__((ext_vector_type(2))) float v2f;
typedef __attribute__((ext_vector_type(8))) float v8f;

#define KTERMS 20   // Taylor terms: ||JA|| ~ 0.6, tau <= ~1.01 -> f32 converged

// --- prep: A = sym(M+M0) -> ws[0..255];  JA = J@A -> ws[256..511] ------------
__global__ __launch_bounds__(256)
void hlf_prep(const float* __restrict__ M, const float* __restrict__ J,
              const float* __restrict__ M0, float* __restrict__ ws) {
    __shared__ float sA[256];
    const int idx = threadIdx.x;
    const int i = idx >> 4, j = idx & 15;
    float a = 0.5f * ((M[i * 16 + j] + M0[i * 16 + j]) +
                      (M[j * 16 + i] + M0[j * 16 + i]));
    sA[idx] = a;
    __syncthreads();
    float acc = 0.f;
#pragma unroll
    for (int k = 0; k < 16; ++k) acc += J[i * 16 + k] * sA[k * 16 + j];
    ws[idx]       = sA[idx];  // A (symmetric)
    ws[256 + idx] = acc;      // JA
}

// --- flow: one wave (32 lanes) per tile of 16 samples ------------------------
__global__ __launch_bounds__(32)
void hlf_flow(const float* __restrict__ x, const float* __restrict__ t,
              const float* __restrict__ ws, float* __restrict__ out) {
    // sU[k][n][i] = ((JA)^k x_n / k!)[i]   (n-major => conflict-free, coalesced)
    __shared__ float sU[KTERMS * 256];
    __shared__ float sA[256];
    __shared__ float sS[16];
    __shared__ float sT[32];

    const int l    = threadIdx.x;      // 0..31, full wave
    const int l15  = l & 15;
    const int half = l >> 4;           // 0 / 1
    const int n0   = blockIdx.x * 16;  // first sample of this tile

    sT[l] = t[l];                      // T == 32 == wave size
#pragma unroll
    for (int q = 0; q < 8; ++q) sA[l + 32 * q] = ws[l + 32 * q];

    // A-operand fragments of JA for the 4 K=4 chunks (reused every iteration).
    // ISA layout (32-bit A 16x4): lane = M, VGPR{0,1} = K{2*half, 2*half+1}.
    const float* JA = ws + 256;
    v2f afrag[4];
#pragma unroll
    for (int c = 0; c < 4; ++c) {
        const int kb = 4 * c + 2 * half;
        afrag[c].x = JA[l15 * 16 + kb];
        afrag[c].y = JA[l15 * 16 + kb + 1];
    }

    // U_0 = x tile (transposed into [n][i]); coalesced 128B global loads.
#pragma unroll
    for (int q = 0; q < 8; ++q) {
        const int p = l + 32 * q;
        const int n = p >> 4, i = p & 15;
        sU[n * 16 + i] = x[(n0 + n) * 16 + i];
    }
    __syncthreads();

    // s_n = 1 + 0.01*tanh(x^T A x); lanes 0..15 each own one sample.
    if (l < 16) {
        float H = 0.f;
        for (int i = 0; i < 16; ++i) {
            const float xi = sU[l * 16 + i];
            float r = 0.f;
#pragma unroll
            for (int j = 0; j < 16; ++j) r += sA[i * 16 + j] * sU[l * 16 + j];
            H += xi * r;
        }
        sS[l] = 1.0f + 0.01f * tanhf(H);
    }
    __syncthreads();

    // Krylov chain: U_k = (JA @ U_{k-1}) / k  via 4x V_WMMA_F32_16X16X4_F32.
    for (int k = 1; k < KTERMS; ++k) {
        const float* Up = sU + (k - 1) * 256;
        v8f d = {};  // zero C accumulator
#pragma unroll
        for (int c = 0; c < 4; ++c) {
            const int kb = 4 * c + 2 * half;
            v2f b;                                   // B 4x16: lane = N, VGPRs = K pair
            b.x = Up[l15 * 16 + kb];
            b.y = Up[l15 * 16 + kb + 1];
            d = __builtin_amdgcn_wmma_f32_16x16x4_f32(
                /*neg_a=*/false, afrag[c], /*neg_b=*/false, b,
                /*c_mod=*/(short)0, d, /*reuse_a=*/false, /*reuse_b=*/false);
        }
        const float rk = 1.0f / (float)k;
        float* Uc = sU + k * 256;
        // C/D 16x16 layout: M = v + 8*half, N = l15
#pragma unroll
        for (int v = 0; v < 8; ++v) {
            const int i = v + 8 * half;
            Uc[l15 * 16 + i] = d[v] * rk;
        }
        __syncthreads();
    }

    // Output: out[t,n,i] = Horner_k( tau^k * U_k[n][i] ), tau = s_n * t.
    // Pair-outer so the K Taylor terms live in registers; t-inner (32 iters).
    // Store pattern per t: lanes cover 2 adjacent samples x 16 dims = 128B.
#pragma unroll 1
    for (int q = 0; q < 8; ++q) {
        const int p = l + 32 * q;
        const int n = p >> 4, i = p & 15;
        float u[KTERMS];
#pragma unroll
        for (int k = 0; k < KTERMS; ++k) u[k] = sU[k * 256 + n * 16 + i];
        const float s = sS[n];
        float* op = out + (size_t)(n0 + n) * 16 + i;
#pragma unroll 1
        for (int tt = 0; tt < 32; ++tt) {
            const float tau = s * sT[tt];
            float acc = u[KTERMS - 1];
#pragma unroll
            for (int k = KTERMS - 2; k >= 0; --k) acc = fmaf(acc, tau, u[k]);
            op[(size_t)tt * 2048 * 16] = acc;
        }
    }
}

extern "C" void kernel_launch(void* const* d_in, const int* in_sizes, int n_in,
                              void* d_out, int out_size, void* d_ws, size_t ws_size,
                              hipStream_t stream) {
    const float* x  = (const float*)d_in[0];  // (2048, 16)
    const float* t  = (const float*)d_in[1];  // (32,)
    const float* M  = (const float*)d_in[2];  // (16, 16)
    const float* J  = (const float*)d_in[3];  // (16, 16)
    const float* M0 = (const float*)d_in[4];  // (16, 16)
    float* out = (float*)d_out;               // (32, 2048, 16)
    float* ws  = (float*)d_ws;                // 512 floats used: A | JA

    hlf_prep<<<1, 256, 0, stream>>>(M, J, M0, ws);
    hlf_flow<<<2048 / 16, 32, 0, stream>>>(x, t, ws, out);
}